// torch_kmeans_37091337568877
// MI455X (gfx1250) — compile-verified
//
#include <hip/hip_runtime.h>

typedef float v2f __attribute__((ext_vector_type(2)));
typedef float v8f __attribute__((ext_vector_type(8)));

#define N_SAMPLES  1000000
#define N_FEATURES 64
#define N_CLUSTERS 128
#define MAX_ITER   5
#define TILE_M     32
#define N_TILES32  (N_SAMPLES / TILE_M)   // 31250 full 32-row tiles
#define ASSIGN_BLOCKS  512
#define ASSIGN_THREADS 256                // 8 waves (wave32)

// ---------------- init: centroids <- init, accumulators <- 0 ----------------
__global__ void km_init(const float* __restrict__ initC, float* cent,
                        float* sums, float* counts, float* inertia) {
  int i = blockIdx.x * blockDim.x + threadIdx.x;
  if (i < N_CLUSTERS * N_FEATURES) { cent[i] = initC[i]; sums[i] = 0.f; }
  if (i < N_CLUSTERS) counts[i] = 0.f;
  if (i == 0) *inertia = 0.f;
}

// ------------- assign: argmin distance + fused segment-sum ------------------
__global__ __launch_bounds__(ASSIGN_THREADS)
void km_assign(const float* __restrict__ X, const float* __restrict__ cent,
               float* __restrict__ sums, float* __restrict__ counts,
               float* __restrict__ inertia, int* __restrict__ labels) {
  __shared__ float s_sums[N_CLUSTERS * N_FEATURES];  // 32 KB
  __shared__ float s_counts[N_CLUSTERS];

  const int tid = threadIdx.x;
  for (int i = tid; i < N_CLUSTERS * N_FEATURES; i += ASSIGN_THREADS) s_sums[i] = 0.f;
  if (tid < N_CLUSTERS) s_counts[tid] = 0.f;
  __syncthreads();

  const int lane  = tid & 31;
  const int wid   = tid >> 5;
  const int gwave = blockIdx.x * (ASSIGN_THREADS / 32) + wid;
  const int lrow  = lane & 15;   // row-within-16 (A layout)
  const int khalf = lane >> 4;   // K-pair selector (A/B layout)
  float inert = 0.f;

  for (int t = gwave; t < N_TILES32; t += ASSIGN_BLOCKS * (ASSIGN_THREADS / 32)) {
    const int rowBase = t * TILE_M;
    const float* xrow0 = X + (size_t)(rowBase + lrow) * N_FEATURES + 2 * khalf;
    const float* xrow1 = xrow0 + 16 * N_FEATURES;

    // A fragments for two 16-row halves; 16 k-steps cover K=64
    v2f a0[16], a1[16];
    float x2p0 = 0.f, x2p1 = 0.f;
#pragma unroll
    for (int s = 0; s < 16; ++s) {
      a0[s] = *(const v2f*)(xrow0 + 4 * s);
      a1[s] = *(const v2f*)(xrow1 + 4 * s);
      x2p0 += a0[s].x * a0[s].x + a0[s].y * a0[s].y;
      x2p1 += a1[s].x * a1[s].x + a1[s].y * a1[s].y;
    }
    const float x2f0 = x2p0 + __shfl_xor(x2p0, 16, 32);  // ||x||^2, row rowBase+lrow
    const float x2f1 = x2p1 + __shfl_xor(x2p1, 16, 32);  // ||x||^2, row rowBase+16+lrow

    float minv0[8], minv1[8];
    int   minn0[8], minn1[8];
#pragma unroll
    for (int r = 0; r < 8; ++r) {
      minv0[r] = 3.4e38f; minn0[r] = 0;
      minv1[r] = 3.4e38f; minn1[r] = 0;
    }

    for (int nt = 0; nt < 8; ++nt) {           // 8 cluster tiles of 16
      const int ncol = nt * 16 + lrow;
      const float* crow = cent + (size_t)ncol * N_FEATURES + 2 * khalf;
      v2f b[8];
      v8f acc0 = {}, acc1 = {};
      float c2p = 0.f;

      // K-half 0: batch-load 8 B fragments, then back-to-back WMMAs
#pragma unroll
      for (int s = 0; s < 8; ++s) b[s] = *(const v2f*)(crow + 4 * s);
#pragma unroll
      for (int s = 0; s < 8; ++s) {
        c2p += b[s].x * b[s].x + b[s].y * b[s].y;
        acc0 = __builtin_amdgcn_wmma_f32_16x16x4_f32(false, a0[s], false, b[s],
                                                     (short)0, acc0, false, false);
        acc1 = __builtin_amdgcn_wmma_f32_16x16x4_f32(false, a1[s], false, b[s],
                                                     (short)0, acc1, false, false);
      }
      // K-half 1
#pragma unroll
      for (int s = 0; s < 8; ++s) b[s] = *(const v2f*)(crow + 32 + 4 * s);
#pragma unroll
      for (int s = 0; s < 8; ++s) {
        c2p += b[s].x * b[s].x + b[s].y * b[s].y;
        acc0 = __builtin_amdgcn_wmma_f32_16x16x4_f32(false, a0[s + 8], false, b[s],
                                                     (short)0, acc0, false, false);
        acc1 = __builtin_amdgcn_wmma_f32_16x16x4_f32(false, a1[s + 8], false, b[s],
                                                     (short)0, acc1, false, false);
      }

      const float c2full = c2p + __shfl_xor(c2p, 16, 32);  // ||c_ncol||^2
#pragma unroll
      for (int r = 0; r < 8; ++r) {            // D: lane holds rows r+8*(l>>4), col l&15
        float sc0 = c2full - 2.f * acc0[r];
        if (sc0 < minv0[r]) { minv0[r] = sc0; minn0[r] = ncol; }
        float sc1 = c2full - 2.f * acc1[r];
        if (sc1 < minv1[r]) { minv1[r] = sc1; minn1[r] = ncol; }
      }
    }

    // reduce over the 16 columns per half-wave (xor stays within each half)
#pragma unroll
    for (int r = 0; r < 8; ++r) {
#pragma unroll
      for (int off = 1; off < 16; off <<= 1) {
        float ov0 = __shfl_xor(minv0[r], off, 32);
        int   on0 = __shfl_xor(minn0[r], off, 32);
        if (ov0 < minv0[r] || (ov0 == minv0[r] && on0 < minn0[r])) { minv0[r] = ov0; minn0[r] = on0; }
        float ov1 = __shfl_xor(minv1[r], off, 32);
        int   on1 = __shfl_xor(minn1[r], off, 32);
        if (ov1 < minv1[r] || (ov1 == minv1[r] && on1 < minn1[r])) { minv1[r] = ov1; minn1[r] = on1; }
      }
    }

    // per-sample epilogue: labels, inertia, fused segment-sum into LDS
#pragma unroll
    for (int m = 0; m < TILE_M; ++m) {
      const int mm  = m & 15;
      const int r   = mm & 7;
      const int src = (mm < 8) ? 0 : 16;       // half-wave holding this row
      int lbl; float mv;
      if (m < 16) { lbl = __shfl(minn0[r], src, 32); mv = __shfl(minv0[r], src, 32); }
      else        { lbl = __shfl(minn1[r], src, 32); mv = __shfl(minv1[r], src, 32); }
      if (lane == m) {                         // lane m holds ||x||^2 of row rowBase+m
        float d2 = mv + ((m < 16) ? x2f0 : x2f1);
        if (d2 < 0.f) d2 = 0.f;
        inert += d2;
        labels[rowBase + m] = lbl;
      }
      // all 32 lanes: add 2 features each (L1-hot reload of the tile row)
      v2f xv = *(const v2f*)(X + (size_t)(rowBase + m) * N_FEATURES + 2 * lane);
      atomicAdd(&s_sums[lbl * N_FEATURES + 2 * lane],     xv.x);
      atomicAdd(&s_sums[lbl * N_FEATURES + 2 * lane + 1], xv.y);
      if (lane == 0) atomicAdd(&s_counts[lbl], 1.f);
    }
  }

  __syncthreads();
  for (int i = tid; i < N_CLUSTERS * N_FEATURES; i += ASSIGN_THREADS) {
    float v = s_sums[i];
    if (v != 0.f) unsafeAtomicAdd(&sums[i], v);
  }
  if (tid < N_CLUSTERS) {
    float v = s_counts[tid];
    if (v != 0.f) unsafeAtomicAdd(&counts[tid], v);
  }
#pragma unroll
  for (int off = 16; off >= 1; off >>= 1) inert += __shfl_xor(inert, off, 32);
  if (lane == 0) unsafeAtomicAdd(inertia, inert);
}

// ------------- update: mean or reseed; re-zero accumulators -----------------
__global__ void km_update(const float* __restrict__ X, const int* __restrict__ reseed,
                          float* cent, float* sums, float* counts,
                          float* inertia, int iter) {
  const int k = blockIdx.x;     // cluster
  const int f = threadIdx.x;    // feature
  const float cnt = counts[k];
  float val;
  if (cnt > 0.f) {
    val = sums[k * N_FEATURES + f] / cnt;
  } else {
    int idx = reseed[iter * N_CLUSTERS + k];
    val = X[(size_t)idx * N_FEATURES + f];
  }
  __syncthreads();  // all reads of counts/sums done before re-zero
  cent[k * N_FEATURES + f] = val;
  sums[k * N_FEATURES + f] = 0.f;
  if (f == 0) counts[k] = 0.f;
  if (k == 0 && f == 0 && iter != MAX_ITER - 1) *inertia = 0.f;  // keep last-iter inertia
}

// ------------- finalize: centroids + inertia -> d_out -----------------------
__global__ void km_finalize(const float* __restrict__ cent,
                            const float* __restrict__ inertia, float* out) {
  int i = blockIdx.x * blockDim.x + threadIdx.x;
  if (i < N_CLUSTERS * N_FEATURES) out[i] = cent[i];
  if (i == 0) out[N_CLUSTERS * N_FEATURES + N_SAMPLES] = *inertia;
}

extern "C" void kernel_launch(void* const* d_in, const int* in_sizes, int n_in,
                              void* d_out, int out_size, void* d_ws, size_t ws_size,
                              hipStream_t stream) {
  (void)in_sizes; (void)n_in; (void)out_size; (void)ws_size;
  const float* X     = (const float*)d_in[0];  // [1M, 64] f32
  const float* initC = (const float*)d_in[1];  // [128, 64] f32
  const int*   rsd   = (const int*)d_in[2];    // [5, 128] i32

  float* out    = (float*)d_out;                         // [0,8192): centroids
  int*   labels = (int*)d_out + N_CLUSTERS * N_FEATURES; // [8192, 8192+1M): labels

  float* cent    = (float*)d_ws;                    // 8192 f32
  float* sums    = cent + N_CLUSTERS * N_FEATURES;  // 8192 f32
  float* counts  = sums + N_CLUSTERS * N_FEATURES;  // 128 f32
  float* inertia = counts + N_CLUSTERS;             // 1 f32

  km_init<<<(N_CLUSTERS * N_FEATURES + 255) / 256, 256, 0, stream>>>(
      initC, cent, sums, counts, inertia);
  for (int it = 0; it < MAX_ITER; ++it) {
    km_assign<<<ASSIGN_BLOCKS, ASSIGN_THREADS, 0, stream>>>(
        X, cent, sums, counts, inertia, labels);
    km_update<<<N_CLUSTERS, N_FEATURES, 0, stream>>>(
        X, rsd, cent, sums, counts, inertia, it);
  }
  km_finalize<<<(N_CLUSTERS * N_FEATURES + 255) / 256, 256, 0, stream>>>(
      cent, inertia, out);
}